// SparseLinear_61761629716950
// MI455X (gfx1250) — compile-verified
//
#include <hip/hip_runtime.h>

typedef __attribute__((ext_vector_type(2))) float v2f;
typedef __attribute__((ext_vector_type(4))) float v4f;
typedef __attribute__((ext_vector_type(8))) float v8f;

#define M_TOK 8192
#define K_DIM 4096
#define N_DIM 4096
#define BK    64
#define BN    64
#define NC    64
#define OCC   4
#define TB    256          // total nonzero blocks
#define BROW  (BN * TB)    // B_data row stride in floats = 16384

#define TM    128          // C tile rows per workgroup
#define LDA   68           // LDS stride for A tile (floats), bank-conflict padding
#define LDB   72           // LDS stride for B tile (floats), bank-conflict padding

__global__ __launch_bounds__(256, 2)
void sparse_linear_wmma_f32(const float* __restrict__ A,
                            const float* __restrict__ Bd,
                            const int*   __restrict__ rowIdx,
                            const int*   __restrict__ dataIdx,
                            float*       __restrict__ C)
{
    __shared__ float lsA[TM * LDA];   // 128 x 64 tile of A (padded)
    __shared__ float lsB[BK * LDB];   // 64 x 64 block of B (padded)

    const int c    = blockIdx.x;      // output column block 0..NC-1
    const int mb   = blockIdx.y;      // row tile 0..M/TM-1
    const int t    = threadIdx.x;
    const int wave = t >> 5;
    const int lane = t & 31;
    const int wm   = wave & 3;        // wave M position (4 waves)
    const int wn   = wave >> 2;       // wave N position (2 waves)
    const int m_base = mb * TM;

    const int half = lane >> 4;       // 0: lanes 0-15, 1: lanes 16-31
    const int l15  = lane & 15;

    v8f acc00 = (v8f)0.f, acc01 = (v8f)0.f, acc10 = (v8f)0.f, acc11 = (v8f)0.f;

    for (int s = 0; s < OCC; ++s) {
        const int r = rowIdx[s * NC + c];
        const int d = dataIdx[s * NC + c];
        const float* __restrict__ Ablk = A  + (size_t)m_base * K_DIM + (size_t)r * BK;
        const float* __restrict__ Bblk = Bd + (size_t)d * BN;

        // ---- stage A tile: 128x64 fp32, 8 x float4 per thread, coalesced ----
        #pragma unroll
        for (int it = 0; it < 8; ++it) {
            int idx = it * 256 + t;
            int row = idx >> 4;              // 0..127
            int c4  = (idx & 15) << 2;       // 0,4,...,60
            v4f v = *(const v4f*)(Ablk + (size_t)row * K_DIM + c4);
            *(v4f*)(&lsA[row * LDA + c4]) = v;
        }
        // ---- stage B block: 64x64 fp32, 4 x float4 per thread ----
        #pragma unroll
        for (int it = 0; it < 4; ++it) {
            int idx = it * 256 + t;
            int row = idx >> 4;              // 0..63
            int c4  = (idx & 15) << 2;
            v4f v = *(const v4f*)(Bblk + (size_t)row * BROW + c4);
            *(v4f*)(&lsB[row * LDB + c4]) = v;
        }
        __syncthreads();

        // ---- prefetch next s-block's A tile toward L2 while we compute ----
        if (s + 1 < OCC) {
            const int rn = rowIdx[(s + 1) * NC + c];
            const float* An = A + (size_t)m_base * K_DIM + (size_t)rn * BK;
            // one byte per lane per row-pair: covers the tile's cachelines
            int row = (t >> 1) & 127;
            __builtin_prefetch(An + (size_t)row * K_DIM + ((t & 1) << 5), 0, 1);
        }

        // ---- 16 WMMA K-steps over this 64-deep K block ----
        #pragma unroll
        for (int k0 = 0; k0 < BK; k0 += 4) {
            const int kk = k0 + half * 2;
            // A fragments (16x4): 2 floats per lane, contiguous -> ds_load_b64
            v2f a0 = *(const v2f*)(&lsA[(wm * 32 +      l15) * LDA + kk]);
            v2f a1 = *(const v2f*)(&lsA[(wm * 32 + 16 + l15) * LDA + kk]);
            // B fragments (4x16): rows kk, kk+1 at this lane's column
            const int cb = wn * 32 + l15;
            v2f b0, b1;
            b0.x = lsB[ kk      * LDB + cb];
            b0.y = lsB[(kk + 1) * LDB + cb];
            b1.x = lsB[ kk      * LDB + cb + 16];
            b1.y = lsB[(kk + 1) * LDB + cb + 16];

            acc00 = __builtin_amdgcn_wmma_f32_16x16x4_f32(false, a0, false, b0,
                                                          (short)0, acc00, false, false);
            acc01 = __builtin_amdgcn_wmma_f32_16x16x4_f32(false, a0, false, b1,
                                                          (short)0, acc01, false, false);
            acc10 = __builtin_amdgcn_wmma_f32_16x16x4_f32(false, a1, false, b0,
                                                          (short)0, acc10, false, false);
            acc11 = __builtin_amdgcn_wmma_f32_16x16x4_f32(false, a1, false, b1,
                                                          (short)0, acc11, false, false);
        }
        __syncthreads();
    }

    // ---- writeback: C 16x16 layout -> lane holds rows (half*8 .. half*8+7) at col l15 ----
    float* __restrict__ Cblk = C + (size_t)m_base * N_DIM + (size_t)c * BN;
    #pragma unroll
    for (int i = 0; i < 2; ++i) {
        const int row0 = wm * 32 + i * 16 + half * 8;
        #pragma unroll
        for (int j = 0; j < 2; ++j) {
            const int col = wn * 32 + j * 16 + l15;
            const v8f accv = (i == 0) ? (j == 0 ? acc00 : acc01)
                                      : (j == 0 ? acc10 : acc11);
            #pragma unroll
            for (int v = 0; v < 8; ++v) {
                Cblk[(size_t)(row0 + v) * N_DIM + col] = accv[v];
            }
        }
    }
}

extern "C" void kernel_launch(void* const* d_in, const int* in_sizes, int n_in,
                              void* d_out, int out_size, void* d_ws, size_t ws_size,
                              hipStream_t stream) {
    const float* A   = (const float*)d_in[0];
    const float* Bd  = (const float*)d_in[1];
    const int*   rI  = (const int*)d_in[2];
    const int*   dI  = (const int*)d_in[3];
    float*       C   = (float*)d_out;

    dim3 grid(NC, M_TOK / TM);   // (64, 64)
    sparse_linear_wmma_f32<<<grid, dim3(256), 0, stream>>>(A, Bd, rI, dI, C);
}